// YoloWrapper_24550033064558
// MI455X (gfx1250) — compile-verified
//
#include <hip/hip_runtime.h>
#include <stdint.h>

// YOLO post-process for MI455X (gfx1250, wave32).
// d_in[0] = preds [16, 84, 8400] f32 ; d_in[1] = imgs (only shape used, 640x640)
// d_out (float) = xyxyn[16*1024*4] ++ conf[16*1024] ++ cls[16*1024] ++ keep[16*1024]

#define NB     16
#define NN     8400
#define NCLS   80
#define KC     1024
#define NPAD   16384          // next pow2 >= NN, for bitonic sort
#define CONF_T 0.25f
#define IOU_T  0.45f
#define IMG_WH 640.0f

typedef float v2f __attribute__((ext_vector_type(2)));
typedef float v8f __attribute__((ext_vector_type(8)));

__device__ __forceinline__ uint32_t sortable_f32(float f) {
  uint32_t u = __float_as_uint(f);
  return (u & 0x80000000u) ? ~u : (u | 0x80000000u);  // monotonic float->u32
}

// ---------------------------------------------------------------- kernel 1 --
// Decode: class max/argmax, cxcywh->xyxy, sort keys (score desc, idx asc).
__global__ void yolo_decode(const float* __restrict__ preds,
                            float* __restrict__ xyxy,
                            float* __restrict__ conf,
                            int*   __restrict__ cls,
                            uint64_t* __restrict__ keys) {
  int gid = blockIdx.x * blockDim.x + threadIdx.x;
  int b = gid / NPAD;
  int n = gid - b * NPAD;
  if (b >= NB) return;
  if (n >= NN) {               // bitonic padding: smallest possible key
    keys[(size_t)b * NPAD + n] = 0ull;
    return;
  }
  const float* p = preds + (size_t)b * (4 + NCLS) * NN;
  __builtin_prefetch(p + (size_t)4 * NN + n, 0, 0);   // global_prefetch_b8

  float cx = p[(size_t)0 * NN + n];
  float cy = p[(size_t)1 * NN + n];
  float bw = p[(size_t)2 * NN + n];
  float bh = p[(size_t)3 * NN + n];

  float best = -1.0f; int bi = 0;
  #pragma unroll 8
  for (int c = 0; c < NCLS; ++c) {
    float s = p[(size_t)(4 + c) * NN + n];
    if (s > best) { best = s; bi = c; }   // strict '>' == jnp.argmax first-max
  }

  size_t o = (size_t)b * NN + n;
  float hw = bw * 0.5f, hh = bh * 0.5f;
  xyxy[o * 4 + 0] = cx - hw;
  xyxy[o * 4 + 1] = cy - hh;
  xyxy[o * 4 + 2] = cx + hw;
  xyxy[o * 4 + 3] = cy + hh;
  conf[o] = best;
  cls[o]  = bi;

  float sc = (best > CONF_T) ? best : -1.0f;          // sub-threshold -> -1
  keys[(size_t)b * NPAD + n] =
      ((uint64_t)sortable_f32(sc) << 32) | (uint32_t)(0xFFFFFFFFu - (uint32_t)n);
}

// ---------------------------------------------------------------- kernel 2 --
// Per-batch in-LDS bitonic sort of 16384 u64 keys (128 KB LDS, CDNA5 WGP LDS).
// Keys are staged with CDNA5 async global->LDS DMA (bypasses VGPRs).
__global__ __launch_bounds__(1024, 1)
void yolo_sort(const uint64_t* __restrict__ keys,
               uint32_t* __restrict__ candidx) {
  __shared__ uint64_t sk[NPAD];                       // 131072 bytes
  const int t = threadIdx.x;
  const int b = blockIdx.x;

  // Async copy 128 KB: 1024 lanes x 8 x b128. LDS offset = low 32 bits of
  // the generic shared-pointer (flat LDS aperture keeps offset in addr[31:0]).
  uint32_t lds0 = (uint32_t)(uintptr_t)(void*)&sk[0];
  uint64_t g0   = (uint64_t)(uintptr_t)(keys + (size_t)b * NPAD);
  #pragma unroll
  for (int r = 0; r < 8; ++r) {
    uint32_t off = (uint32_t)((r * 1024 + t) * 16);
    uint32_t la  = lds0 + off;
    uint64_t ga  = g0 + off;
    asm volatile("global_load_async_to_lds_b128 %0, %1, off"
                 :: "v"(la), "v"(ga) : "memory");
  }
  asm volatile("s_wait_asynccnt 0" ::: "memory");
  __syncthreads();

  // Bitonic sort, descending. 16 elements per thread, pairs guarded by ixj>i.
  for (int k = 2; k <= NPAD; k <<= 1) {
    for (int j = k >> 1; j > 0; j >>= 1) {
      #pragma unroll
      for (int r = 0; r < NPAD / 1024; ++r) {
        int i   = t + r * 1024;
        int ixj = i ^ j;
        if (ixj > i) {
          uint64_t a = sk[i], c = sk[ixj];
          bool desc = ((i & k) == 0);
          if ((a < c) == desc) { sk[i] = c; sk[ixj] = a; }
        }
      }
      __syncthreads();
    }
  }

  // Top-K indices: n = 0xFFFFFFFF - low32(key)
  candidx[(size_t)b * KC + t] = 0xFFFFFFFFu - (uint32_t)sk[t];
}

// ---------------------------------------------------------------- kernel 3 --
// Two-phase greedy NMS, one workgroup (32 waves) per batch.
// Phase 1 (no barriers): build full 1024x1024 suppression bitmask M in LDS.
//   Each wave owns a 32-column j-stripe (two 16-wide WMMA tiles). The IoU
//   denominator term area_i + area_j is produced as a 16x16 rank-2 outer-sum
//   by V_WMMA_F32_16X16X4_F32: A[M][k]={area_i,1,0,0}, B[k][N]={1;area_j;0;0}.
//   D layout (lane L -> N=L%16, M=v+8*(L>=16)) hands each lane 8 (i,j)
//   denominators per tile; inter is VALU. The threshold test is done
//   division-free: iou > T  <=>  inter > T*(area_sum - inter + eps)
//   (valid since all areas/unions are positive). Per-row bits come from
//   wave32 ballots (low half = row v, high half = row v+8).
// Phase 2 (1 barrier): wave 0 alone runs the serial greedy scan over M.
__global__ __launch_bounds__(1024, 1)
void yolo_nms(const uint32_t* __restrict__ candidx,
              const float* __restrict__ xyxy,
              const float* __restrict__ conf,
              const int*   __restrict__ cls,
              float* __restrict__ out) {
  __shared__ uint32_t M[KC][32];       // 128 KB suppression bitmask (row i, word j/32)
  __shared__ float4   sbox[KC];        // 16 KB
  __shared__ float    sarea[KC];       // 4 KB
  __shared__ uint32_t skeep[KC / 32];  // 128 B
  const int t = threadIdx.x;
  const int b = blockIdx.x;
  const int lane = t & 31, wv = t >> 5;
  const bool lo = lane < 16;

  uint32_t ci = candidx[(size_t)b * KC + t];
  if (ci >= NN) ci = NN - 1;           // padding can never win top-K; safety
  size_t src = (size_t)b * NN + ci;

  float4 bx = *(const float4*)(xyxy + src * 4);
  float  cj = conf[src];
  int    kj = cls[src];

  sbox[t] = bx;
  sarea[t] = (bx.z - bx.x) * (bx.w - bx.y);
  __syncthreads();

  // ---- Phase 1: suppression bitmask, wave-private j-stripe [32*wv, 32*wv+32)
  const int jA = (wv << 5) + (lane & 15);
  const int jB = jA + 16;
  const float4 boxA = sbox[jA];
  const float4 boxB = sbox[jB];

  // B operand (loop-invariant): lanes 0-15 carry K0=1.0 / K1=area_j; 16-31 zero.
  v2f BvA, BvB;
  BvA.x = lo ? 1.0f : 0.0f;
  BvA.y = lo ? sarea[(wv << 5) + lane] : 0.0f;
  BvB.x = lo ? 1.0f : 0.0f;
  BvB.y = lo ? sarea[(wv << 5) + 16 + lane] : 0.0f;

  for (int blk = 0; blk < KC; blk += 16) {
    // A operand: lanes 0-15 carry K0=area_i(row=lane) / K1=1.0; 16-31 zero.
    v2f Av;
    Av.x = lo ? sarea[blk + lane] : 0.0f;
    Av.y = lo ? 1.0f : 0.0f;
    v8f zc = {};
    v8f DA = __builtin_amdgcn_wmma_f32_16x16x4_f32(
        false, Av, false, BvA, (short)0, zc, false, false);
    v8f DB = __builtin_amdgcn_wmma_f32_16x16x4_f32(
        false, Av, false, BvB, (short)0, zc, false, false);

    const int mibase = blk + (lo ? 0 : 8);
    #pragma unroll
    for (int v = 0; v < 8; ++v) {
      const int mi = mibase + v;       // row index handled by this lane-half
      float4 bi = sbox[mi];
      // tile A: division-free IoU threshold test
      float lx = fmaxf(boxA.x, bi.x), ly = fmaxf(boxA.y, bi.y);
      float rx = fminf(boxA.z, bi.z), ry = fminf(boxA.w, bi.w);
      float inter = fmaxf(rx - lx, 0.0f) * fmaxf(ry - ly, 0.0f);
      float unionA = DA[v] - inter + 1e-9f;          // DA[v] = area_i + area_jA
      bool supA = (inter > IOU_T * unionA) && (jA > mi);
      uint32_t balA = (uint32_t)__ballot(supA);
      // tile B
      lx = fmaxf(boxB.x, bi.x); ly = fmaxf(boxB.y, bi.y);
      rx = fminf(boxB.z, bi.z); ry = fminf(boxB.w, bi.w);
      inter = fmaxf(rx - lx, 0.0f) * fmaxf(ry - ly, 0.0f);
      float unionB = DB[v] - inter + 1e-9f;          // DB[v] = area_i + area_jB
      bool supB = (inter > IOU_T * unionB) && (jB > mi);
      uint32_t balB = (uint32_t)__ballot(supB);
      if (lane == 0) {
        M[blk + v][wv]     = (balA & 0xFFFFu) | ((balB & 0xFFFFu) << 16);
        M[blk + v + 8][wv] = (balA >> 16)     | (balB & 0xFFFF0000u);
      }
    }
  }
  __syncthreads();

  // ---- Phase 2: serial greedy scan on wave 0 only (no barriers inside).
  if (wv == 0) {
    uint32_t kw = 0xFFFFFFFFu;                 // lane owns keep bits [32*lane..)
    for (int i = 0; i < KC; ++i) {
      uint32_t w = __shfl(kw, i >> 5, 32);     // broadcast word holding bit i
      if ((w >> (i & 31)) & 1u)                // box i kept -> apply its row
        kw &= ~M[i][lane];
    }
    skeep[lane] = kw;
  }
  __syncthreads();

  // ---- Outputs
  bool keep = (((skeep[t >> 5] >> (t & 31)) & 1u) != 0u) && (cj > CONF_T);
  const float inv = 1.0f / IMG_WH;
  float* oxy = out;
  float* ocf = out + (size_t)NB * KC * 4;
  float* ocl = ocf + (size_t)NB * KC;
  float* okp = ocl + (size_t)NB * KC;
  size_t oo = (size_t)b * KC + t;
  oxy[oo * 4 + 0] = keep ? bx.x * inv : 0.0f;
  oxy[oo * 4 + 1] = keep ? bx.y * inv : 0.0f;
  oxy[oo * 4 + 2] = keep ? bx.z * inv : 0.0f;
  oxy[oo * 4 + 3] = keep ? bx.w * inv : 0.0f;
  ocf[oo] = keep ? cj : 0.0f;
  ocl[oo] = keep ? (float)kj : -1.0f;
  okp[oo] = keep ? 1.0f : 0.0f;
}

// ------------------------------------------------------------------- launch --
extern "C" void kernel_launch(void* const* d_in, const int* in_sizes, int n_in,
                              void* d_out, int out_size, void* d_ws, size_t ws_size,
                              hipStream_t stream) {
  const float* preds = (const float*)d_in[0];
  // d_in[1] (imgs) only contributes its fixed 640x640 shape.

  char* ws = (char*)d_ws;
  // workspace layout (16B-aligned offsets)
  uint64_t* keys    = (uint64_t*)(ws);                               // 2,097,152 B
  float*    xyxy    = (float*)   (ws + 2097152);                     // 2,150,400 B
  float*    conf    = (float*)   (ws + 2097152 + 2150400);           //   537,600 B
  int*      cls     = (int*)     (ws + 2097152 + 2150400 + 537600);  //   537,600 B
  uint32_t* candidx = (uint32_t*)(ws + 2097152 + 2150400 + 1075200); //    65,536 B

  yolo_decode<<<(NB * NPAD) / 256, 256, 0, stream>>>(preds, xyxy, conf, cls, keys);
  yolo_sort<<<NB, 1024, 0, stream>>>(keys, candidx);
  yolo_nms<<<NB, 1024, 0, stream>>>(candidx, xyxy, conf, cls, (float*)d_out);
}